// OntoEncoder_89361089561007
// MI455X (gfx1250) — compile-verified
//
#include <hip/hip_runtime.h>
#include <hip/hip_bf16.h>

// ---------------------------------------------------------------------------
// OntoEncoder (DAG autoencoder) for MI455X / gfx1250.
// Memory-bound (~0.1 GFLOP vs ~120 MB => ~5us HBM floor @ 23.3 TB/s).
// Pipeline: stats(x) -> h1 -> stats(h1) -> h2 -> stats(h2) -> fused decoder.
// CDNA5 paths used:
//   * global_load_async_to_lds_b128/b32 + s_wait_asynccnt for LDS row staging
//   * v_wmma_f32_16x16x4_f32 with ones-A for batch column-sum reductions
// ---------------------------------------------------------------------------

#define G_  8192
#define T1_ 1024
#define T2_ 256
#define T3_ 64
#define B_  1024
#define EPS_ 1e-5f
#define RSPLIT_ 8                 // row-blocks per stats pass (disjoint partials)
#define ROWS_PER_BLK_ (B_ / RSPLIT_)   // 128

typedef __attribute__((ext_vector_type(2))) float v2f;
typedef __attribute__((ext_vector_type(8))) float v8f;

// Generic pointer to LDS: low 32 bits of the flat address ARE the DS byte
// offset (ISA 10.2: the LDS aperture is picked by the high 32 bits,
// the LDS address is the low 32 bits).
__device__ __forceinline__ unsigned lds_lo32(const void* p) {
    return (unsigned)(unsigned long long)(size_t)p;
}
__device__ __forceinline__ void async_copy_b128(void* lds_dst, const void* gsrc) {
    unsigned l = lds_lo32(lds_dst);
    asm volatile("global_load_async_to_lds_b128 %0, %1, off"
                 :: "v"(l), "v"(gsrc) : "memory");
}
__device__ __forceinline__ void async_copy_b32(void* lds_dst, const void* gsrc) {
    unsigned l = lds_lo32(lds_dst);
    asm volatile("global_load_async_to_lds_b32 %0, %1, off"
                 :: "v"(l), "v"(gsrc) : "memory");
}
__device__ __forceinline__ void wait_async0() {
    asm volatile("s_wait_asynccnt 0x0" ::: "memory");
}

// ---------------------------------------------------------------------------
// Column sums / sums-of-squares via WMMA f32 16x16x4 with an all-ones A.
// With ones-A, D[m,n] = sum_k B[k,n] (+C) regardless of the hardware's K-slot
// mapping; only the documented N = lane%16 B/D mapping is relied on.
// Each wave handles 16 columns; 8 waves/block -> 128 columns per block.
// grid.y row-blocks write disjoint partial slots (no atomics, deterministic).
// Fixed 128-row trip count; pointer-increment addressing (no u64 mul in loop).
// ---------------------------------------------------------------------------
__global__ __launch_bounds__(256)
void k_stats_partial(const float* __restrict__ X, int ncols,
                     float* __restrict__ psum, float* __restrict__ psumsq) {
    const int lane = threadIdx.x & 31;
    const int wave = threadIdx.x >> 5;
    const int col  = blockIdx.x * 128 + wave * 16 + (lane & 15);
    const int rowoff = (lane >> 4) * 2;   // lanes 0-15: rows {0,1}; 16-31: {2,3}

    const size_t step = (size_t)4 * (size_t)ncols;
    const float* p0 = X + (size_t)(blockIdx.y * ROWS_PER_BLK_ + rowoff) * ncols + col;
    const float* p1 = p0 + ncols;

    v2f a; a[0] = 1.0f; a[1] = 1.0f;      // ones => layout-proof column sum
    v8f cs = {}; v8f cq = {};
#pragma unroll 4
    for (int r = 0; r < ROWS_PER_BLK_; r += 4) {
        float x0 = *p0;
        float x1 = *p1;
        p0 += step; p1 += step;
        v2f bm; bm[0] = x0;      bm[1] = x1;
        v2f bq; bq[0] = x0 * x0; bq[1] = x1 * x1;
        cs = __builtin_amdgcn_wmma_f32_16x16x4_f32(false, a, false, bm, (short)0, cs, false, false);
        cq = __builtin_amdgcn_wmma_f32_16x16x4_f32(false, a, false, bq, (short)0, cq, false, false);
    }
    if (lane < 16) {   // lanes 16-31 hold duplicate rows (M=8..15 of ones-product)
        psum  [(size_t)blockIdx.y * ncols + col] = cs[0];
        psumsq[(size_t)blockIdx.y * ncols + col] = cq[0];
    }
}

__global__ __launch_bounds__(256)
void k_stats_finalize(const float* __restrict__ psum, const float* __restrict__ psumsq,
                      int n, float invB, float* __restrict__ mu, float* __restrict__ rsig) {
    int i = blockIdx.x * 256 + threadIdx.x;
    if (i < n) {
        float s = 0.0f, q = 0.0f;
#pragma unroll
        for (int j = 0; j < RSPLIT_; ++j) {   // fixed order: deterministic
            s += psum  [(size_t)j * n + i];
            q += psumsq[(size_t)j * n + i];
        }
        float m = s * invB;
        float v = q * invB - m * m;
        mu[i]   = m;
        rsig[i] = rsqrtf(v + EPS_);
    }
}

// ---------------------------------------------------------------------------
// Encoder level 1: h1[b,t] = relu(sum_k xn[b, gi[t,k]] * W1e[t,k] + b1e[t])
// One block per batch row; x row (32KB) staged into LDS via async b128,
// normalized in place, then gathered. Index/weight reads are b128.
// ---------------------------------------------------------------------------
__global__ __launch_bounds__(256)
void k_enc1(const float* __restrict__ X, const float* __restrict__ mu,
            const float* __restrict__ rsig, const float* __restrict__ W1e,
            const float* __restrict__ b1e, const int* __restrict__ gidx,
            float* __restrict__ h1) {
    __shared__ float sx[G_];
    const int b = blockIdx.x;
    const float* xrow = X + (size_t)b * G_;
#pragma unroll
    for (int it = 0; it < 8; ++it) {
        int e = (it * 256 + threadIdx.x) * 4;
        async_copy_b128(&sx[e], xrow + e);
    }
    wait_async0();
    __syncthreads();
    for (int e = threadIdx.x; e < G_; e += 256)
        sx[e] = (sx[e] - mu[e]) * rsig[e];
    __syncthreads();
    for (int t = threadIdx.x; t < T1_; t += 256) {
        const int4*   gi4 = (const int4*)  (gidx + t * 16);
        const float4* w4  = (const float4*)(W1e  + t * 16);
        float acc = b1e[t];
#pragma unroll
        for (int q = 0; q < 4; ++q) {
            int4   gi = gi4[q];
            float4 w  = w4[q];
            acc += sx[gi.x] * w.x + sx[gi.y] * w.y + sx[gi.z] * w.z + sx[gi.w] * w.w;
        }
        h1[(size_t)b * T1_ + t] = fmaxf(acc, 0.0f);
    }
}

// ---------------------------------------------------------------------------
// Encoder level 2: h2[b,t] = relu(sum_k h1n[b, ti[t,k]] * W2e[t,k] + b2e[t])
// ---------------------------------------------------------------------------
__global__ __launch_bounds__(256)
void k_enc2(const float* __restrict__ H1, const float* __restrict__ mu,
            const float* __restrict__ rsig, const float* __restrict__ W2e,
            const float* __restrict__ b2e, const int* __restrict__ tidx,
            float* __restrict__ h2) {
    __shared__ float sh[T1_];
    const int b = blockIdx.x;
    const float* row = H1 + (size_t)b * T1_;
    {
        int e = threadIdx.x * 4;              // 256 threads * 16B = 4KB row
        async_copy_b128(&sh[e], row + e);
    }
    wait_async0();
    __syncthreads();
    for (int e = threadIdx.x; e < T1_; e += 256)
        sh[e] = (sh[e] - mu[e]) * rsig[e];
    __syncthreads();
    const int t = threadIdx.x;                // exactly T2_=256 terms
    const int4*   ti4 = (const int4*)  (tidx + t * 8);
    const float4* w4  = (const float4*)(W2e  + t * 8);
    float acc = b2e[t];
#pragma unroll
    for (int q = 0; q < 2; ++q) {
        int4   ti = ti4[q];
        float4 w  = w4[q];
        acc += sh[ti.x] * w.x + sh[ti.y] * w.y + sh[ti.z] * w.z + sh[ti.w] * w.w;
    }
    h2[(size_t)b * T2_ + t] = fmaxf(acc, 0.0f);
}

// ---------------------------------------------------------------------------
// Fused decoder: z -> d2 -> d1 -> out, entirely row-local in LDS.
// par/W tables read as b128; output written as b128 (4 genes per thread).
// ---------------------------------------------------------------------------
__global__ __launch_bounds__(256)
void k_dec(const float* __restrict__ H2, const float* __restrict__ mu2,
           const float* __restrict__ rsig2,
           const float* __restrict__ W3e, const float* __restrict__ b3e,
           const int* __restrict__ tidx3,
           const float* __restrict__ W2d, const float* __restrict__ b2d,
           const int* __restrict__ par3,
           const float* __restrict__ W1d, const float* __restrict__ b1d,
           const int* __restrict__ par2,
           const float* __restrict__ Wgd, const float* __restrict__ bgd,
           const int* __restrict__ parg,
           float* __restrict__ out) {
    __shared__ float sh2[T2_];
    __shared__ float sz[T3_];
    __shared__ float sd2[T2_];
    __shared__ float sd1[T1_];
    const int b   = blockIdx.x;
    const int tid = threadIdx.x;

    async_copy_b32(&sh2[tid], H2 + (size_t)b * T2_ + tid);
    wait_async0();
    __syncthreads();
    sh2[tid] = (sh2[tid] - mu2[tid]) * rsig2[tid];
    __syncthreads();

    if (tid < T3_) {                               // latent z
        const int4*   ti4 = (const int4*)  (tidx3 + tid * 8);
        const float4* w4  = (const float4*)(W3e   + tid * 8);
        float acc = b3e[tid];
#pragma unroll
        for (int q = 0; q < 2; ++q) {
            int4   ti = ti4[q];
            float4 w  = w4[q];
            acc += sh2[ti.x] * w.x + sh2[ti.y] * w.y + sh2[ti.z] * w.z + sh2[ti.w] * w.w;
        }
        sz[tid] = fmaxf(acc, 0.0f);
    }
    __syncthreads();

    {                                              // d2 (256 terms, 2 parents)
        int2   pp = ((const int2*)par3)[tid];
        float2 ww = ((const float2*)W2d)[tid];
        float acc = b2d[tid] + sz[pp.x] * ww.x + sz[pp.y] * ww.y;
        sd2[tid] = fmaxf(acc, 0.0f);
    }
    __syncthreads();

    {                                              // d1: 4 consecutive terms/thread
        const int t0 = tid * 4;
        int4   pa = ((const int4*)(par2 + t0 * 2))[0];
        int4   pb = ((const int4*)(par2 + t0 * 2))[1];
        float4 wa = ((const float4*)(W1d + t0 * 2))[0];
        float4 wb = ((const float4*)(W1d + t0 * 2))[1];
        float4 bb = ((const float4*)(b1d + t0))[0];
        float4 o;
        o.x = fmaxf(bb.x + sd2[pa.x] * wa.x + sd2[pa.y] * wa.y, 0.0f);
        o.y = fmaxf(bb.y + sd2[pa.z] * wa.z + sd2[pa.w] * wa.w, 0.0f);
        o.z = fmaxf(bb.z + sd2[pb.x] * wb.x + sd2[pb.y] * wb.y, 0.0f);
        o.w = fmaxf(bb.w + sd2[pb.z] * wb.z + sd2[pb.w] * wb.w, 0.0f);
        *(float4*)&sd1[t0] = o;
    }
    __syncthreads();

    float* orow = out + (size_t)b * G_;
#pragma unroll
    for (int it = 0; it < 8; ++it) {               // 4 consecutive genes/thread
        const int g0 = it * 1024 + tid * 4;
        int4   pa = ((const int4*)(parg + g0 * 2))[0];
        int4   pb = ((const int4*)(parg + g0 * 2))[1];
        float4 wa = ((const float4*)(Wgd + g0 * 2))[0];
        float4 wb = ((const float4*)(Wgd + g0 * 2))[1];
        float4 bb = ((const float4*)(bgd + g0))[0];
        float4 o;
        o.x = bb.x + sd1[pa.x] * wa.x + sd1[pa.y] * wa.y;
        o.y = bb.y + sd1[pa.z] * wa.z + sd1[pa.w] * wa.w;
        o.z = bb.z + sd1[pb.x] * wb.x + sd1[pb.y] * wb.y;
        o.w = bb.w + sd1[pb.z] * wb.z + sd1[pb.w] * wb.w;
        *(float4*)(orow + g0) = o;
    }
}

// ---------------------------------------------------------------------------
extern "C" void kernel_launch(void* const* d_in, const int* in_sizes, int n_in,
                              void* d_out, int out_size, void* d_ws, size_t ws_size,
                              hipStream_t stream) {
    const float* x    = (const float*)d_in[0];
    const float* W1e  = (const float*)d_in[1];
    const float* b1e  = (const float*)d_in[2];
    const float* W2e  = (const float*)d_in[3];
    const float* b2e  = (const float*)d_in[4];
    const float* W3e  = (const float*)d_in[5];
    const float* b3e  = (const float*)d_in[6];
    const float* W2d  = (const float*)d_in[7];
    const float* b2d  = (const float*)d_in[8];
    const float* W1d  = (const float*)d_in[9];
    const float* b1d  = (const float*)d_in[10];
    const float* Wgd  = (const float*)d_in[11];
    const float* bgd  = (const float*)d_in[12];
    const int* gidx1  = (const int*)d_in[13];
    const int* tidx2  = (const int*)d_in[14];
    const int* tidx3  = (const int*)d_in[15];
    const int* par3   = (const int*)d_in[16];
    const int* par2   = (const int*)d_in[17];
    const int* parg   = (const int*)d_in[18];
    float* out = (float*)d_out;
    float* ws  = (float*)d_ws;

    // Workspace layout (floats); all offsets multiples of 4 (16B aligned).
    float* psum   = ws;                    // RSPLIT_ * 8192 = 65536
    float* psumsq = ws + 65536;            // 65536
    float* mux    = ws + 131072;           // 8192
    float* rsx    = ws + 139264;           // 8192
    float* mu1    = ws + 147456;           // 1024
    float* rs1    = ws + 148480;           // 1024
    float* mu2    = ws + 149504;           // 256
    float* rs2    = ws + 149760;           // 256
    float* h1     = ws + 150016;           // 1024*1024
    float* h2     = ws + 1198592;          // 1024*256   (total ~5.85 MB)

    const dim3 blk(256);
    const float invB = 1.0f / (float)B_;

    // stats(x) -> enc1 -> stats(h1) -> enc2 -> stats(h2) -> fused decoder
    k_stats_partial <<<dim3(G_ / 128, RSPLIT_), blk, 0, stream>>>(x, G_, psum, psumsq);
    k_stats_finalize<<<G_ / 256, blk, 0, stream>>>(psum, psumsq, G_, invB, mux, rsx);

    k_enc1<<<B_, blk, 0, stream>>>(x, mux, rsx, W1e, b1e, gidx1, h1);

    k_stats_partial <<<dim3(T1_ / 128, RSPLIT_), blk, 0, stream>>>(h1, T1_, psum, psumsq);
    k_stats_finalize<<<T1_ / 256, blk, 0, stream>>>(psum, psumsq, T1_, invB, mu1, rs1);

    k_enc2<<<B_, blk, 0, stream>>>(h1, mu1, rs1, W2e, b2e, tidx2, h2);

    k_stats_partial <<<dim3(T2_ / 128, RSPLIT_), blk, 0, stream>>>(h2, T2_, psum, psumsq);
    k_stats_finalize<<<1, blk, 0, stream>>>(psum, psumsq, T2_, invB, mu2, rs2);

    k_dec<<<B_, blk, 0, stream>>>(h2, mu2, rs2, W3e, b3e, tidx3,
                                  W2d, b2d, par3, W1d, b1d, par2,
                                  Wgd, bgd, parg, out);
}